// DeformableFeatureAggregation_14173392077261
// MI455X (gfx1250) — compile-verified
//
#include <hip/hip_runtime.h>
#include <math.h>

// ---------------- problem constants ----------------
#define EMBED 256
#define NGRP  8
#define GRPD  32
#define NLVL  4
#define NCAM  6
#define NPTS  13
#define NLRN  6
#define NANCH 900
#define BS    2
#define NTOT  (BS*NANCH)                 // 1800
#define WCOLS (NGRP*NCAM*NLVL*NPTS)      // 2496 (used slice of wfc output)
#define WFC_LD (NGRP*20*NLVL*NPTS)       // 8320 (full wfc_w column count)
#define NSAMP (NCAM*NLVL*NPTS)           // 312
#define PIX_TOTAL 14960                  // sum of H*W over levels

__device__ __constant__ int   d_FMH[NLVL]  = {64, 32, 16, 8};
__device__ __constant__ int   d_FMW[NLVL]  = {176, 88, 44, 22};
__device__ __constant__ int   d_LBASE[NLVL]= {0, 11264, 14080, 14784};
__device__ __constant__ float d_FIX[7][3]  = {
    {0.f,0.f,0.f},{0.45f,0.f,0.f},{-0.45f,0.f,0.f},{0.f,0.45f,0.f},
    {0.f,-0.45f,0.f},{0.f,0.f,0.45f},{0.f,0.f,-0.45f}};

typedef float v2f __attribute__((ext_vector_type(2)));
typedef float v8f __attribute__((ext_vector_type(8)));

// CDNA5 async memory->LDS copy (VGLOBAL GLOBAL_LOAD_ASYNC_TO_LDS_B128,
// tracked by ASYNCcnt). lds_dst must be 16B aligned; per-lane EXEC masking ok.
__device__ __forceinline__ void async_g2l_b128(void* lds_dst, const void* gsrc)
{
    unsigned lds = (unsigned)(uintptr_t)lds_dst;                // addrspace(3) offset
    unsigned long long g = (unsigned long long)(uintptr_t)gsrc;
    asm volatile("global_load_async_to_lds_b128 %0, %1, off"
                 :: "v"(lds), "v"(g) : "memory");
}
__device__ __forceinline__ void wait_async0()
{
    asm volatile("s_wait_asynccnt 0x0" ::: "memory");
}

// ---------------------------------------------------------------------------
// Kernel 0: channel-major -> pixel-major transpose of one feature level.
//   in : fm  [12][256][HW]      out: fmT [12][PIX_TOTAL][256] (level at lbase)
// Stage-in uses async global->LDS b128 (one 16B packet per thread = 32x32
// tile); stride-36 row pitch keeps 16B alignment and spreads banks.
// ---------------------------------------------------------------------------
__global__ void k_transpose(const float* __restrict__ fm, float* __restrict__ fmT,
                            int HW, int lbase)
{
    __shared__ float tile[32][36];       // [channel][pixel], 144B row pitch
    const int tid = threadIdx.x;         // 256
    const int px0 = blockIdx.x * 32;
    const int ch0 = blockIdx.y * 32;
    const int bc  = blockIdx.z;          // 0..11

    // phase 1: async copy 32 channels x 32 pixels into LDS (4 px per lane)
    {
        const int chl = tid >> 3;        // 0..31
        const int qx  = tid & 7;         // 0..7 -> pixels qx*4..qx*4+3
        const int px  = px0 + qx * 4;
        if (px < HW) {                   // HW % 16 == 0 -> full 4-px packet valid
            const float* src = fm + ((long)bc * EMBED + ch0 + chl) * (long)HW + px;
            async_g2l_b128(&tile[chl][qx * 4], src);
        }
    }
    wait_async0();
    __syncthreads();

    // phase 2: coalesced write-out, channels contiguous per pixel row
    const int tx = tid & 31;             // channel within tile
    const int ty = tid >> 5;             // 0..7
    #pragma unroll
    for (int i = 0; i < 4; ++i) {
        int pxl = ty + i * 8;
        int px  = px0 + pxl;
        if (px < HW)
            fmT[((long)bc * PIX_TOTAL + lbase + px) * EMBED + ch0 + tx] = tile[tx][pxl];
    }
}

// ---------------------------------------------------------------------------
// Kernel 1: per-anchor geometry: learnable keypoints (sigmoid of 18-dot GEMV),
// box-frame keypoints, yaw rotation, camera projection -> proj[NTOT][78][2]
// ---------------------------------------------------------------------------
__global__ void k_points(const float* __restrict__ inst,   // [NTOT][256]
                         const float* __restrict__ anchor, // [NTOT][11]
                         const float* __restrict__ kps_w,  // [256][18]
                         const float* __restrict__ kps_b,  // [18]
                         const float* __restrict__ projm,  // [BS][6][4][4]
                         const float* __restrict__ imwh,   // [BS][6][2]
                         float* __restrict__ proj)         // [NTOT][78][2]
{
    __shared__ float sFeat[EMBED];
    __shared__ float sAnc[11];
    __shared__ float sLs[NLRN * 3];
    __shared__ float sKp[NPTS][3];

    const int t    = threadIdx.x;        // 128 threads
    const int aIdx = blockIdx.x;         // 0..1799
    const int b    = aIdx / NANCH;

    for (int i = t; i < EMBED; i += 128) sFeat[i] = inst[(long)aIdx * EMBED + i];
    if (t < 11) sAnc[t] = anchor[(long)aIdx * 11 + t];
    __syncthreads();

    if (t < NLRN * 3) {                  // 18 sigmoid dot-products (K=256)
        float s = kps_b[t];
        for (int k = 0; k < EMBED; ++k) s += sFeat[k] * kps_w[k * (NLRN * 3) + t];
        sLs[t] = 1.f / (1.f + __expf(-s)) - 0.5f;
    }
    __syncthreads();

    if (t < NPTS) {
        float sz0 = __expf(sAnc[3]), sz1 = __expf(sAnc[4]), sz2 = __expf(sAnc[5]);
        float vx, vy, vz;
        if (t < 7) { vx = d_FIX[t][0]*sz0; vy = d_FIX[t][1]*sz1; vz = d_FIX[t][2]*sz2; }
        else { int j = t - 7; vx = sLs[j*3+0]*sz0; vy = sLs[j*3+1]*sz1; vz = sLs[j*3+2]*sz2; }
        float sy = sAnc[6], cy = sAnc[7];
        sKp[t][0] = cy * vx - sy * vy + sAnc[0];
        sKp[t][1] = sy * vx + cy * vy + sAnc[1];
        sKp[t][2] = vz + sAnc[2];
    }
    __syncthreads();

    if (t < NCAM * NPTS) {               // 78 (cam, point) projections
        int c = t / NPTS, p = t % NPTS;
        const float* P = projm + ((long)(b * NCAM + c)) * 16;
        float kx = sKp[p][0], ky = sKp[p][1], kz = sKp[p][2];
        float x2 = P[0]*kx + P[1]*ky + P[2]*kz  + P[3];
        float y2 = P[4]*kx + P[5]*ky + P[6]*kz  + P[7];
        float zz = P[8]*kx + P[9]*ky + P[10]*kz + P[11];
        float zi = 1.f / fmaxf(zz, 1e-5f);
        float u = x2 * zi / imwh[(b * NCAM + c) * 2 + 0];
        float v = y2 * zi / imwh[(b * NCAM + c) * 2 + 1];
        proj[((long)aIdx * 78 + t) * 2 + 0] = u;
        proj[((long)aIdx * 78 + t) * 2 + 1] = v;
    }
}

// ---------------------------------------------------------------------------
// Kernel 2: f32 WMMA GEMM  C[M,N] = (A [+ A2]) @ B + bias [+ Res]
// 256 threads = 8 waves; each wave -> one 16x16 tile, K-loop of 64
// v_wmma_f32_16x16x4_f32. A tile staged in LDS (padded stride 260 floats =
// 1040B rows: bank-conflict-free ds_load_2addr_b64 fragments, 16B aligned for
// the async stage-in path used when A2 == nullptr).
// A-frag: lanes 0-15 M=lane K={0,1}; lanes 16-31 same M, K={2,3}.
// B-frag mirrors with N across lanes (B row-major, stride ldb).
// C/D: vgpr r -> row r (lanes 0-15) / r+8 (lanes 16-31), col = lane&15.
// ---------------------------------------------------------------------------
__global__ void k_gemm_wmma(const float* __restrict__ A,
                            const float* __restrict__ A2,   // nullable add
                            const float* __restrict__ Bm, int ldb,
                            const float* __restrict__ bias,
                            const float* __restrict__ Res,  // nullable residual (N==256)
                            float* __restrict__ C, int M, int N, int ldc)
{
    __shared__ float sA[16 * 260];
    const int tid  = threadIdx.x;
    const int wave = tid >> 5;
    const int lane = tid & 31;
    const int m0   = blockIdx.x * 16;
    const int n0   = blockIdx.y * 128 + wave * 16;

    if (A2) {
        // fused A + A2 stage-in (wfc GEMM)
        for (int i = tid; i < 16 * EMBED; i += 256) {
            int r = i >> 8, k = i & 255;
            int row = m0 + r; if (row >= M) row = M - 1;
            sA[r * 260 + k] = A[(long)row * EMBED + k] + A2[(long)row * EMBED + k];
        }
    } else {
        // pure copy: async global->LDS b128, one float4 per thread x 4 iters
        for (int i = tid; i < 16 * (EMBED / 4); i += 256) {   // 1024 packets
            int r  = i >> 6;            // row 0..15
            int k4 = (i & 63) * 4;      // col 0,4,..252
            int row = m0 + r; if (row >= M) row = M - 1;
            async_g2l_b128(&sA[r * 260 + k4], A + (long)row * EMBED + k4);
        }
        wait_async0();
    }
    __syncthreads();

    const int lm = lane & 15;
    const int kh = lane >> 4;            // 0/1
    v8f acc = {};
    if (n0 < N) {                        // wave-uniform guard: EXEC stays all-1s
        const float* bp = Bm + n0 + lm;
        #pragma unroll 8
        for (int kk = 0; kk < EMBED; kk += 4) {
            int kb = kk + kh * 2;
            v2f a = *(const v2f*)&sA[lm * 260 + kb];      // ds_load_b64
            v2f bv;
            bv.x = bp[(long)kb * ldb];
            bv.y = bp[(long)(kb + 1) * ldb];
            acc = __builtin_amdgcn_wmma_f32_16x16x4_f32(
                      false, a, false, bv, (short)0, acc, false, false);
        }
        const int col = n0 + lm;
        const float bb = bias ? bias[col] : 0.f;
        #pragma unroll
        for (int r = 0; r < 8; ++r) {
            int row = m0 + r + kh * 8;
            if (row < M) {
                float v = acc[r] + bb;
                if (Res) v += Res[(long)row * EMBED + col];
                C[(long)row * ldc + col] = v;
            }
        }
    }
}

// ---------------------------------------------------------------------------
// Kernel 3: per-anchor softmax (8 groups over 312 logits) + bilinear gather
// from pixel-major fmT + group-weighted fusion -> fused[NTOT][256].
// Block = 128 threads = 2 anchor slots x 64 threads; thread owns 4 channels
// (float4 corner loads, 1KB contiguous per corner per slot). Sample tables
// (clamped corner offsets + zeroed OOB bilinear weights) built once in LDS.
// ---------------------------------------------------------------------------
#define APB 2
__global__ void k_sample_fuse(const float* __restrict__ fmT,
                              const float* __restrict__ wraw,  // [NTOT][2496]
                              const float* __restrict__ proj,  // [NTOT][78][2]
                              float* __restrict__ fused)       // [NTOT][256]
{
    __shared__ float sW[APB][WCOLS];         // softmaxed weights   19968 B
    __shared__ int   sOff[APB][NSAMP * 4];   // corner float-offsets 9984 B
    __shared__ float sCw[APB][NSAMP * 4];    // corner bilinear wts  9984 B
    __shared__ float sRed[APB][16];

    const int tid  = threadIdx.x;            // 0..127
    const int slot = tid >> 6;               // 0..1
    const int t    = tid & 63;               // 0..63
    const int aIdx = blockIdx.x * APB + slot;
    const int b    = aIdx / NANCH;

    // 1) stage raw logits
    for (int i = t; i < WCOLS; i += 64) sW[slot][i] = wraw[(long)aIdx * WCOLS + i];
    __syncthreads();

    // 2) softmax per group (layout idx = s*8 + g); 8 lanes per group, shfl_xor
    {
        const int g = t >> 3, j = t & 7;
        float mx = -3.4e38f;
        for (int s = j; s < NSAMP; s += 8) mx = fmaxf(mx, sW[slot][s * 8 + g]);
        for (int d = 4; d >= 1; d >>= 1) mx = fmaxf(mx, __shfl_xor(mx, d, 32));
        float sm = 0.f;
        for (int s = j; s < NSAMP; s += 8) sm += __expf(sW[slot][s * 8 + g] - mx);
        for (int d = 4; d >= 1; d >>= 1) sm += __shfl_xor(sm, d, 32);
        if (j == 0) { sRed[slot][g * 2] = mx; sRed[slot][g * 2 + 1] = 1.f / sm; }
    }
    __syncthreads();
    for (int i = t; i < WCOLS; i += 64) {
        int g = i & 7;
        sW[slot][i] = __expf(sW[slot][i] - sRed[slot][g * 2]) * sRed[slot][g * 2 + 1];
    }

    // 3) sample table: grid_sample(align_corners=False, zeros pad), wh == 1
    for (int s = t; s < NSAMP; s += 64) {
        int c = s / (NLVL * NPTS);
        int l = (s / NPTS) % NLVL;
        int p = s % NPTS;
        float u = proj[((long)aIdx * 78 + c * NPTS + p) * 2 + 0];
        float v = proj[((long)aIdx * 78 + c * NPTS + p) * 2 + 1];
        int Wl = d_FMW[l], Hl = d_FMH[l];
        float x = u * (float)Wl - 0.5f;
        float y = v * (float)Hl - 0.5f;
        float xf = floorf(x), yf = floorf(y);
        float fx = x - xf, fy = y - yf;
        int x0 = (int)xf, y0 = (int)yf;
        int planeBase = ((b * NCAM + c) * PIX_TOTAL + d_LBASE[l]) * EMBED;
        float cw[4] = {(1.f - fx) * (1.f - fy), fx * (1.f - fy),
                       (1.f - fx) * fy,         fx * fy};
        #pragma unroll
        for (int k = 0; k < 4; ++k) {
            int xi = x0 + (k & 1), yi = y0 + (k >> 1);
            bool ok = (xi >= 0) & (xi < Wl) & (yi >= 0) & (yi < Hl);
            int xc = min(max(xi, 0), Wl - 1);
            int yc = min(max(yi, 0), Hl - 1);
            sOff[slot][s * 4 + k] = planeBase + (yc * Wl + xc) * EMBED;
            sCw[slot][s * 4 + k]  = ok ? cw[k] : 0.f;
        }
    }
    __syncthreads();

    // 4) gather + fuse: thread owns channels [d0, d0+4), group = d0/32
    const int d0 = t * 4;
    const int g  = d0 >> 5;
    float4 acc = make_float4(0.f, 0.f, 0.f, 0.f);
    for (int s = 0; s < NSAMP; ++s) {
        int sn = (s + 1 < NSAMP) ? s + 1 : s;
        __builtin_prefetch(fmT + sOff[slot][sn * 4] + d0, 0, 0);
        float wa = sW[slot][s * 8 + g];
        float4 f = make_float4(0.f, 0.f, 0.f, 0.f);
        #pragma unroll
        for (int k = 0; k < 4; ++k) {
            float c2 = sCw[slot][s * 4 + k];
            const float4 vv = *(const float4*)(fmT + sOff[slot][s * 4 + k] + d0);
            f.x += c2 * vv.x; f.y += c2 * vv.y; f.z += c2 * vv.z; f.w += c2 * vv.w;
        }
        acc.x += wa * f.x; acc.y += wa * f.y; acc.z += wa * f.z; acc.w += wa * f.w;
    }
    *(float4*)(fused + (long)aIdx * EMBED + d0) = acc;
}

// ---------------------------------------------------------------------------
extern "C" void kernel_launch(void* const* d_in, const int* in_sizes, int n_in,
                              void* d_out, int out_size, void* d_ws, size_t ws_size,
                              hipStream_t stream)
{
    (void)in_sizes; (void)n_in; (void)out_size; (void)ws_size;
    const float* inst   = (const float*)d_in[0];
    const float* anchor = (const float*)d_in[1];
    const float* aembed = (const float*)d_in[2];
    const float* projm  = (const float*)d_in[3];
    const float* imwh   = (const float*)d_in[4];
    const float* kps_w  = (const float*)d_in[5];
    const float* kps_b  = (const float*)d_in[6];
    const float* wfc_w  = (const float*)d_in[7];
    const float* wfc_b  = (const float*)d_in[8];
    const float* out_w  = (const float*)d_in[9];
    const float* out_b  = (const float*)d_in[10];
    const float* fm[NLVL] = {(const float*)d_in[11], (const float*)d_in[12],
                             (const float*)d_in[13], (const float*)d_in[14]};
    float* out = (float*)d_out;

    // workspace layout (floats)
    float* ws = (float*)d_ws;
    const size_t fmT_n   = (size_t)12 * PIX_TOTAL * EMBED;   // 45,957,120
    const size_t wraw_n  = (size_t)NTOT * WCOLS;             //  4,492,800
    const size_t proj_n  = (size_t)NTOT * 78 * 2;            //    280,800
    float* fmT   = ws;
    float* wraw  = fmT + fmT_n;
    float* projb = wraw + wraw_n;
    float* fused = projb + proj_n;

    // 0) transpose each level to pixel-major (async global->LDS stage-in)
    const int HWs[NLVL]   = {11264, 2816, 704, 176};
    const int LBAS[NLVL]  = {0, 11264, 14080, 14784};
    for (int l = 0; l < NLVL; ++l) {
        dim3 g((HWs[l] + 31) / 32, EMBED / 32, 12);
        k_transpose<<<g, dim3(256), 0, stream>>>(fm[l], fmT, HWs[l], LBAS[l]);
    }

    // 1) geometry + projection
    k_points<<<dim3(NTOT), dim3(128), 0, stream>>>(inst, anchor, kps_w, kps_b,
                                                   projm, imwh, projb);

    // 2) wfc GEMM (only the used 2496-col slice of the 8320-col weight)
    {
        dim3 g((NTOT + 15) / 16, (WCOLS + 127) / 128);
        k_gemm_wmma<<<g, dim3(256), 0, stream>>>(inst, aembed, wfc_w, WFC_LD,
                                                 wfc_b, nullptr, wraw,
                                                 NTOT, WCOLS, WCOLS);
    }

    // 3) softmax + bilinear gather + group fusion
    k_sample_fuse<<<dim3(NTOT / APB), dim3(64 * APB), 0, stream>>>(fmT, wraw,
                                                                   projb, fused);

    // 4) output GEMM + bias + residual (async A stage-in path)
    {
        dim3 g((NTOT + 15) / 16, (EMBED + 127) / 128);
        k_gemm_wmma<<<g, dim3(256), 0, stream>>>(fused, nullptr, out_w, EMBED,
                                                 out_b, inst, out,
                                                 NTOT, EMBED, EMBED);
    }
}